// GCN_34033320853729
// MI455X (gfx1250) — compile-verified
//
#include <hip/hip_runtime.h>
#include <hip/hip_bf16.h>

typedef __attribute__((ext_vector_type(2))) float v2f;
typedef __attribute__((ext_vector_type(8))) float v8f;

// ---------------- degree / normalization ----------------

__global__ void gcn_init_deg(float* deg, int N) {
    int i = blockIdx.x * blockDim.x + threadIdx.x;
    if (i < N) deg[i] = 1.0f;   // self-loop
}

__global__ void gcn_accum_deg(const long long* __restrict__ col, float* deg, int E) {
    int e = blockIdx.x * blockDim.x + threadIdx.x;
    if (e < E) atomicAdd(&deg[(int)col[e]], 1.0f);
}

__global__ void gcn_finalize_dinv(float* deg, int N) {
    int i = blockIdx.x * blockDim.x + threadIdx.x;
    if (i < N) {
        float d = deg[i];
        deg[i] = (d > 0.0f) ? (1.0f / sqrtf(d)) : 0.0f;
    }
}

// ---------------- WMMA f32 GEMM: C[N,16] = A[N,K] @ B[K,16] ----------------
// One wave per 16-row tile. Uses V_WMMA_F32_16X16X4_F32 (exact f32 path).
// A (16x4 f32) lane layout: lanes 0-15 hold M=lane, lanes 16-31 hold M=lane-16;
// VGPR v holds K = v + 2*(lane>>4).  B (4x16): VGPR v holds K = v + 2*half,
// N = lane&15.  C (16x16): VGPR v holds M = v + 8*half, N = lane&15.

__global__ void gcn_gemm_wmma_f32(const float* __restrict__ A,
                                  const float* __restrict__ B,
                                  float* __restrict__ C,
                                  int K, int ntiles) {
    int wave = threadIdx.x >> 5;
    int tile = blockIdx.x * (blockDim.x >> 5) + wave;
    if (tile >= ntiles) return;          // wave-uniform guard: EXEC stays all-1s
    int lane = threadIdx.x & 31;
    int m    = lane & 15;
    int half = lane >> 4;

    const float* arow = A + (size_t)(tile * 16 + m) * K;
    v8f acc = {};
    for (int k = 0; k < K; k += 4) {
        int ka = k + 2 * half;
        v2f a, b;
        a.x = arow[ka];
        a.y = arow[ka + 1];
        b.x = B[(size_t)ka * 16 + m];
        b.y = B[(size_t)(ka + 1) * 16 + m];
        acc = __builtin_amdgcn_wmma_f32_16x16x4_f32(
            /*neg_a=*/false, a, /*neg_b=*/false, b,
            /*c_mod=*/(short)0, acc, /*reuse_a=*/false, /*reuse_b=*/false);
    }
    float* crow = C + (size_t)tile * 16 * 16 + m;   // N = m, column offset
#pragma unroll
    for (int v = 0; v < 8; ++v) {
        int mm = v + 8 * half;
        crow[(size_t)mm * 16] = acc[v];
    }
}

// ---------------- aggregation ----------------

// agg[j][f] = dinv[j]^2 * h[j][f]   (self-loop contribution; also initializes agg)
__global__ void gcn_selfloop_init(const float* __restrict__ h,
                                  const float* __restrict__ dinv,
                                  float* __restrict__ agg, int N) {
    int t = blockIdx.x * blockDim.x + threadIdx.x;
    if (t < N * 16) {
        int j = t >> 4;
        float di = dinv[j];
        agg[t] = di * di * h[t];
    }
}

// one thread handles 4 features of one edge (float4 gather, 4 atomics into L2)
__global__ void gcn_edge_agg(const long long* __restrict__ row,
                             const long long* __restrict__ col,
                             const float* __restrict__ dinv,
                             const float* __restrict__ h,
                             float* __restrict__ agg, int E) {
    int t = blockIdx.x * blockDim.x + threadIdx.x;
    if (t >= E * 4) return;
    int e = t >> 2;
    int q = (t & 3) * 4;
    int r = (int)row[e];
    int c = (int)col[e];
    float norm = dinv[r] * dinv[c];
    const float4 hv = *(const float4*)(h + (size_t)r * 16 + q);
    float* dst = agg + (size_t)c * 16 + q;
    atomicAdd(dst + 0, norm * hv.x);
    atomicAdd(dst + 1, norm * hv.y);
    atomicAdd(dst + 2, norm * hv.z);
    atomicAdd(dst + 3, norm * hv.w);
}

__global__ void gcn_bias_act(float* __restrict__ agg,
                             const float* __restrict__ bias,
                             int N, int do_relu) {
    int t = blockIdx.x * blockDim.x + threadIdx.x;
    if (t < N * 16) {
        float v = agg[t] + bias[t & 15];
        if (do_relu) v = fmaxf(v, 0.0f);
        agg[t] = v;
    }
}

// ---------------- driver ----------------

extern "C" void kernel_launch(void* const* d_in, const int* in_sizes, int n_in,
                              void* d_out, int out_size, void* d_ws, size_t ws_size,
                              hipStream_t stream) {
    const float*     x  = (const float*)d_in[0];
    const long long* ei = (const long long*)d_in[1];   // [2, E] int64
    const float*     W1 = (const float*)d_in[2];
    const float*     b1 = (const float*)d_in[3];
    const float*     W2 = (const float*)d_in[4];
    const float*     b2 = (const float*)d_in[5];
    float* out = (float*)d_out;

    const int N = in_sizes[0] / 128;
    const int E = in_sizes[1] / 2;
    const long long* row = ei;
    const long long* col = ei + E;

    float* dinv = (float*)d_ws;              // N floats
    float* hbuf = dinv + N;                  // 16N floats (h1, then h2)
    float* agg  = hbuf + (size_t)16 * N;     // 16N floats (agg1 -> relu'd h)

    const int ntiles = (N + 15) / 16;        // N = 100000 -> 6250 exact
    const int TPB = 256;

    // normalization coefficients
    gcn_init_deg<<<(N + TPB - 1) / TPB, TPB, 0, stream>>>(dinv, N);
    gcn_accum_deg<<<(E + TPB - 1) / TPB, TPB, 0, stream>>>(col, dinv, E);
    gcn_finalize_dinv<<<(N + TPB - 1) / TPB, TPB, 0, stream>>>(dinv, N);

    // ---- layer 1 ----
    {
        int waves_per_blk = 4;                         // 128 threads = 4 waves
        int blocks = (ntiles + waves_per_blk - 1) / waves_per_blk;
        gcn_gemm_wmma_f32<<<blocks, 32 * waves_per_blk, 0, stream>>>(x, W1, hbuf, 128, ntiles);
    }
    gcn_selfloop_init<<<(N * 16 + TPB - 1) / TPB, TPB, 0, stream>>>(hbuf, dinv, agg, N);
    gcn_edge_agg<<<(E * 4 + TPB - 1) / TPB, TPB, 0, stream>>>(row, col, dinv, hbuf, agg, E);
    gcn_bias_act<<<(N * 16 + TPB - 1) / TPB, TPB, 0, stream>>>(agg, b1, N, /*relu=*/1);

    // ---- layer 2 ----  (h2 reuses hbuf; final agg goes straight to d_out)
    {
        int waves_per_blk = 4;
        int blocks = (ntiles + waves_per_blk - 1) / waves_per_blk;
        gcn_gemm_wmma_f32<<<blocks, 32 * waves_per_blk, 0, stream>>>(agg, W2, hbuf, 16, ntiles);
    }
    gcn_selfloop_init<<<(N * 16 + TPB - 1) / TPB, TPB, 0, stream>>>(hbuf, dinv, out, N);
    gcn_edge_agg<<<(E * 4 + TPB - 1) / TPB, TPB, 0, stream>>>(row, col, dinv, hbuf, out, E);
    gcn_bias_act<<<(N * 16 + TPB - 1) / TPB, TPB, 0, stream>>>(out, b2, N, /*relu=*/0);
}